// DynamicGraphConstructor_695784702508
// MI455X (gfx1250) — compile-verified
//
#include <hip/hip_runtime.h>
#include <hip/hip_bf16.h>
#include <math.h>

typedef float v2f __attribute__((ext_vector_type(2)));
typedef float v8f __attribute__((ext_vector_type(8)));

#define N_NODES 2048
#define T_STEPS 512
#define H_DIM   256
#define E_DIM   128          // H/2
#define TOPK    10
#define E_FIXED 65536
#define E_TOTAL (E_FIXED + N_NODES * TOPK)   // 86016

// ---------------------------------------------------------------------------
// Kernel 1: x_agg[n][h] = mean_t x[n][t][h].  HBM-bound: 1.07 GB streamed once.
// 64 lanes per row, float4 (b128) coalesced loads: 512 B/wave/iter contiguous.
// ---------------------------------------------------------------------------
__global__ void k_mean_x(const float* __restrict__ x, float* __restrict__ xa) {
    const int n  = blockIdx.x;           // 0..2047
    const int t4 = threadIdx.x;          // 0..63 -> h = 4*t4 .. 4*t4+3
    const float4* xv = (const float4*)(x + (size_t)n * T_STEPS * H_DIM);
    float4 acc = make_float4(0.f, 0.f, 0.f, 0.f);
#pragma unroll 4
    for (int t = 0; t < T_STEPS; ++t) {
        float4 v = xv[(size_t)t * (H_DIM / 4) + t4];
        acc.x += v.x; acc.y += v.y; acc.z += v.z; acc.w += v.w;
    }
    const float s = 1.0f / (float)T_STEPS;
    float4* o = (float4*)(xa + (size_t)n * H_DIM);
    o[t4] = make_float4(acc.x * s, acc.y * s, acc.z * s, acc.w * s);
}

// ---------------------------------------------------------------------------
// Kernel 2: e = x_agg @ W^T  (2048x256 @ 256x128 -> 2048x128), fp32 WMMA.
// One wave per 16x16 output tile, K stepped by 4 via V_WMMA_F32_16X16X4_F32.
// A-layout (16x4 f32): lane<16 -> K={0,1}; lane>=16 -> K={2,3}, M = lane&15.
// B-layout (4x16 f32): VGPR v: lanes 0-15 row K=v, lanes 16-31 row K=v+2.
// ---------------------------------------------------------------------------
__global__ void k_gemm_e(const float* __restrict__ xa,
                         const float* __restrict__ W,
                         float* __restrict__ e) {
    const int tile = blockIdx.x;            // 0..1023
    const int tm   = tile >> 3;             // 0..127 (row tile)
    const int tn   = tile & 7;              // 0..7   (col tile)
    const int lane = threadIdx.x;           // 0..31
    const int half = lane >> 4;             // 0/1
    const int l16  = lane & 15;

    const int m = tm * 16 + l16;            // A row for this lane
    const int n = tn * 16 + l16;            // B col (= W row) for this lane
    v8f acc = {};
    for (int k0 = 0; k0 < H_DIM; k0 += 4) {
        const int kk = k0 + 2 * half;
        v2f a, b;
        a.x = xa[(size_t)m * H_DIM + kk + 0];
        a.y = xa[(size_t)m * H_DIM + kk + 1];
        // B[k][n] = W[n][k]
        b.x = W[(size_t)n * H_DIM + kk + 0];
        b.y = W[(size_t)n * H_DIM + kk + 1];
        acc = __builtin_amdgcn_wmma_f32_16x16x4_f32(
            false, a, false, b, (short)0, acc, false, false);
    }
    // D layout: VGPR v -> row (v + 8*half), col = l16
    const int mbase = tm * 16 + half * 8;
    const int nbase = tn * 16;
#pragma unroll
    for (int v = 0; v < 8; ++v)
        e[(size_t)(mbase + v) * E_DIM + nbase + l16] = acc[v];
}

// ---------------------------------------------------------------------------
// Kernel 3: row-normalize e (per row of 128).
// ---------------------------------------------------------------------------
__global__ void k_normalize_e(float* __restrict__ e) {
    const int n = blockIdx.x;
    const int h = threadIdx.x;      // 0..127
    __shared__ float red[128];
    float v = e[(size_t)n * E_DIM + h];
    red[h] = v * v;
    __syncthreads();
    for (int s = 64; s > 0; s >>= 1) {
        if (h < s) red[h] += red[h + s];
        __syncthreads();
    }
    float nrm = fmaxf(sqrtf(red[0]), 1e-12f);
    e[(size_t)n * E_DIM + h] = v / nrm;
}

// ---------------------------------------------------------------------------
// Kernel 4: sim = e @ e^T  (2048x2048, K=128), fp32 WMMA, one wave per tile.
// e is 1 MB -> lives in L2; 16 MB sim also L2-resident on the 192 MB cache.
// ---------------------------------------------------------------------------
__global__ void k_gemm_sim(const float* __restrict__ e, float* __restrict__ sim) {
    const int tile = blockIdx.x;            // 0..16383
    const int tm   = tile >> 7;             // 0..127
    const int tn   = tile & 127;            // 0..127
    const int lane = threadIdx.x;
    const int half = lane >> 4;
    const int l16  = lane & 15;

    const int m = tm * 16 + l16;
    const int n = tn * 16 + l16;
    v8f acc = {};
    for (int k0 = 0; k0 < E_DIM; k0 += 4) {
        const int kk = k0 + 2 * half;
        v2f a, b;
        a.x = e[(size_t)m * E_DIM + kk + 0];
        a.y = e[(size_t)m * E_DIM + kk + 1];
        // B[k][n] = e[n][k]  (e @ e^T)
        b.x = e[(size_t)n * E_DIM + kk + 0];
        b.y = e[(size_t)n * E_DIM + kk + 1];
        acc = __builtin_amdgcn_wmma_f32_16x16x4_f32(
            false, a, false, b, (short)0, acc, false, false);
    }
    const int mbase = tm * 16 + half * 8;
    const int nbase = tn * 16;
#pragma unroll
    for (int v = 0; v < 8; ++v)
        sim[(size_t)(mbase + v) * N_NODES + nbase + l16] = acc[v];
}

// ---------------------------------------------------------------------------
// Kernel 5: per-row top-(K+1), drop first (self), write dynamic edges.
// Row (8 KB) is LDS-resident; 11 argmax passes, ties -> lower index
// (matches jax.lax.top_k ordering). Output rows as float (harness float* out).
// out layout: [0..E_TOTAL) = src row, [E_TOTAL..2E) = dst row, [2E..3E) = attr.
// ---------------------------------------------------------------------------
__global__ void k_topk(const float* __restrict__ sim,
                       const float* __restrict__ mix_logit,
                       float* __restrict__ out) {
    const int n   = blockIdx.x;
    const int tid = threadIdx.x;    // 0..255
    __shared__ float svals[N_NODES];
    __shared__ float rv[256];
    __shared__ int   ri[256];

    for (int i = tid; i < N_NODES; i += 256)
        svals[i] = sim[(size_t)n * N_NODES + i];
    __syncthreads();

    const float alpha = 1.0f / (1.0f + expf(-mix_logit[0]));

    for (int sel = 0; sel < TOPK + 1; ++sel) {
        float bv = -INFINITY; int bi = 0;
        for (int i = tid; i < N_NODES; i += 256) {
            float v = svals[i];
            if (v > bv) { bv = v; bi = i; }   // strict > keeps lowest index
        }
        rv[tid] = bv; ri[tid] = bi;
        __syncthreads();
        for (int s = 128; s > 0; s >>= 1) {
            if (tid < s) {
                if (rv[tid + s] > rv[tid] ||
                    (rv[tid + s] == rv[tid] && ri[tid + s] < ri[tid])) {
                    rv[tid] = rv[tid + s]; ri[tid] = ri[tid + s];
                }
            }
            __syncthreads();
        }
        if (tid == 0) {
            if (sel > 0) {
                const int epos = E_FIXED + n * TOPK + (sel - 1);
                out[epos]               = (float)n;        // src
                out[E_TOTAL + epos]     = (float)ri[0];    // dst
                out[2 * E_TOTAL + epos] = rv[0] * alpha;   // attr
            }
            svals[ri[0]] = -INFINITY;
        }
        __syncthreads();
    }
}

// ---------------------------------------------------------------------------
// Kernel 6: copy fixed edges, attrs scaled by (1 - alpha).
// ---------------------------------------------------------------------------
__global__ void k_fixed(const long long* __restrict__ fei,
                        const float* __restrict__ fea,
                        const float* __restrict__ mix_logit,
                        float* __restrict__ out) {
    const int i = blockIdx.x * blockDim.x + threadIdx.x;
    if (i >= E_FIXED) return;
    const float alpha = 1.0f / (1.0f + expf(-mix_logit[0]));
    out[i]               = (float)fei[i];               // src row
    out[E_TOTAL + i]     = (float)fei[E_FIXED + i];     // dst row
    out[2 * E_TOTAL + i] = fea[i] * (1.0f - alpha);     // attr
}

// ---------------------------------------------------------------------------
extern "C" void kernel_launch(void* const* d_in, const int* in_sizes, int n_in,
                              void* d_out, int out_size, void* d_ws, size_t ws_size,
                              hipStream_t stream) {
    const float*     x    = (const float*)d_in[0];
    const long long* fei  = (const long long*)d_in[1];   // int64 edge index
    const float*     fea  = (const float*)d_in[2];
    const float*     W    = (const float*)d_in[3];
    const float*     mixl = (const float*)d_in[4];
    float* out = (float*)d_out;

    // workspace: xa (2 MB) | e (1 MB) | sim (16 MB)
    float* xa  = (float*)d_ws;
    float* e   = xa + (size_t)N_NODES * H_DIM;
    float* sim = e  + (size_t)N_NODES * E_DIM;

    k_mean_x     <<<N_NODES, 64, 0, stream>>>(x, xa);
    k_gemm_e     <<<(N_NODES / 16) * (E_DIM / 16), 32, 0, stream>>>(xa, W, e);
    k_normalize_e<<<N_NODES, 128, 0, stream>>>(e);
    k_gemm_sim   <<<(N_NODES / 16) * (N_NODES / 16), 32, 0, stream>>>(e, sim);
    k_topk       <<<N_NODES, 256, 0, stream>>>(sim, mixl, out);
    k_fixed      <<<(E_FIXED + 255) / 256, 256, 0, stream>>>(fei, fea, mixl, out);
}